// RoutingCapsule_37469294691081
// MI455X (gfx1250) — compile-verified
//
#include <hip/hip_runtime.h>

typedef __attribute__((ext_vector_type(2))) float v2f;
typedef __attribute__((ext_vector_type(8))) float v8f;

#define Bsz   16
#define Isz   2048
#define Psz   16
#define Osz   64
#define Dsz   32
#define ODsz  2048   // O*D
#define IBLK  32     // i-values per GEMM block
#define ICHUNK 64    // i-values per routing block

// ---------------------------------------------------------------- utilities
__global__ __launch_bounds__(256) void zero_kernel(float* __restrict__ p, int n) {
    int i = blockIdx.x * 256 + threadIdx.x;
    if (i < n) p[i] = 0.0f;
}

// v = squash(scale * sacc + bias_od); also re-zero sacc for the next pass.
__global__ __launch_bounds__(256) void squash_kernel(
    float* __restrict__ sacc, const float* __restrict__ bias,
    float* __restrict__ dst, float scale)
{
    int idx = blockIdx.x * 256 + threadIdx.x;        // < Bsz*ODsz = 32768
    float s  = sacc[idx] * scale + bias[idx & (ODsz - 1)];
    float sq = s * s;
    dst[idx]  = (sq / (1.0f + sq)) * s * __frsqrt_rn(sq + 1e-9f);
    sacc[idx] = 0.0f;
}

// ---------------------------------------------------------- u_hat via WMMA
// u_hat[b,i,od] = sum_p x[b,i,p] * W[i,od,p]
// Per i: A = x[:,i,:] (16x16, M=b, K=p), B = W[i]^T (16 x 2048, K=p, N=od).
// One wave: 4 tiles of 16 od-columns; K=16 via 4 chained v_wmma_f32_16x16x4_f32.
// Also register-accumulates sum_i u_hat over its IBLK i's -> atomic flush (c0=1/64 path).
__global__ __launch_bounds__(256) void uhat_gemm_kernel(
    const float* __restrict__ x, const float* __restrict__ W,
    float* __restrict__ uhat, float* __restrict__ sacc)
{
    const int lane = threadIdx.x & 31;
    const int wave = threadIdx.x >> 5;
    const int lrow = lane & 15;      // M (=b) for A; N (=od offset) for B/D
    const int lhi  = lane >> 4;      // selects K pair {0,1} vs {2,3} / D row half
    const int odw  = blockIdx.x * 512 + wave * 64;   // wave's first od column
    const int i0   = blockIdx.y * IBLK;

    v8f sum[4];
    #pragma unroll
    for (int t = 0; t < 4; ++t) sum[t] = (v8f)0.0f;

    for (int ii = 0; ii < IBLK; ++ii) {
        const int i = i0 + ii;
        // A: lane holds x[b=lrow, i, p = 4*ks + 2*lhi + {0,1}]  (ISA 16x4 f32 A layout)
        const float* xp = x + (size_t)lrow * (Isz * Psz) + (size_t)i * Psz + 2 * lhi;
        v2f a[4];
        #pragma unroll
        for (int ks = 0; ks < 4; ++ks) a[ks] = *(const v2f*)(xp + 4 * ks);

        const float* wp = W + (size_t)i * (ODsz * Psz);
        #pragma unroll
        for (int t = 0; t < 4; ++t) {
            const int odc = odw + t * 16 + lrow;
            // B: lane holds W[i, od=odc, p = 4*ks + 2*lhi + {0,1}] (K rows striped like A)
            const float* wq = wp + (size_t)odc * Psz + 2 * lhi;
            v8f d = (v8f)0.0f;
            #pragma unroll
            for (int ks = 0; ks < 4; ++ks) {
                v2f bv = *(const v2f*)(wq + 4 * ks);
                d = __builtin_amdgcn_wmma_f32_16x16x4_f32(
                        false, a[ks], false, bv, (short)0, d, false, false);
            }
            sum[t] += d;
            // D layout: VGPR r, lanes 0-15 -> (M=r, N), lanes 16-31 -> (M=r+8, N)
            const size_t ubase = (size_t)i * ODsz + odc;
            #pragma unroll
            for (int r = 0; r < 8; ++r) {
                const int M = r + 8 * lhi;     // batch index b
                uhat[(size_t)M * (Isz * (size_t)ODsz) + ubase] = d[r];
            }
        }
    }
    // flush partial sum_i u_hat into global s accumulator
    #pragma unroll
    for (int t = 0; t < 4; ++t) {
        const int od = odw + t * 16 + lrow;
        #pragma unroll
        for (int r = 0; r < 8; ++r) {
            const int M = r + 8 * lhi;
            atomicAdd(&sacc[M * ODsz + od], sum[t][r]);
        }
    }
}

// ------------------------------------------------------------ routing pass
// For each (b,i): blog[o] = (iter==2 ? b1 : 0) + sum_d u_hat[o,d]*v[o,d];
// c = softmax_o(blog); accumulate c*u_hat into s. Thread t owns od = 8t..8t+7
// (one o per thread, o = t>>2, shared by 4 lanes).
__global__ __launch_bounds__(256) void routing_pass_kernel(
    const float* __restrict__ uhat, const float* __restrict__ vprev,
    const float* __restrict__ bprev, float* __restrict__ bnew,
    float* __restrict__ sacc, int iter)
{
    const int t   = threadIdx.x;
    const int b   = blockIdx.y;
    const int i0  = blockIdx.x * ICHUNK;
    const int od0 = t * 8;
    const int o   = t >> 2;
    const int wv  = t >> 5;

    __shared__ float red[16];

    float vv[8];
    {
        const float* vp = vprev + b * ODsz + od0;
        float4 va = *(const float4*)vp;
        float4 vb = *(const float4*)(vp + 4);
        vv[0]=va.x; vv[1]=va.y; vv[2]=va.z; vv[3]=va.w;
        vv[4]=vb.x; vv[5]=vb.y; vv[6]=vb.z; vv[7]=vb.w;
    }
    float acc[8];
    #pragma unroll
    for (int j = 0; j < 8; ++j) acc[j] = 0.0f;

    for (int ii = 0; ii < ICHUNK; ++ii) {
        const int i = i0 + ii;
        const float* up = uhat + ((size_t)b * Isz + i) * ODsz + od0;
        float u[8];
        {
            float4 ua = *(const float4*)up;
            float4 ub = *(const float4*)(up + 4);
            u[0]=ua.x; u[1]=ua.y; u[2]=ua.z; u[3]=ua.w;
            u[4]=ub.x; u[5]=ub.y; u[6]=ub.z; u[7]=ub.w;
        }
        // per-o dot(u_hat, v): partial over this thread's 8 d's, reduce 4 lanes
        float p = 0.0f;
        #pragma unroll
        for (int j = 0; j < 8; ++j) p += u[j] * vv[j];
        p += __shfl_xor(p, 1, 32);
        p += __shfl_xor(p, 2, 32);
        float blog = p;
        if (iter == 2) blog += bprev[((size_t)b * Isz + i) * Osz + o];
        if (iter == 1 && (t & 3) == 0) bnew[((size_t)b * Isz + i) * Osz + o] = blog;

        // softmax over o = 0..63 (8 distinct o per wave, x4 replication in lane bits 0-1)
        float m = blog;
        m = fmaxf(m, __shfl_xor(m, 4, 32));
        m = fmaxf(m, __shfl_xor(m, 8, 32));
        m = fmaxf(m, __shfl_xor(m, 16, 32));
        if ((t & 31) == 0) red[wv] = m;
        __syncthreads();
        float gm = red[0];
        #pragma unroll
        for (int w = 1; w < 8; ++w) gm = fmaxf(gm, red[w]);
        __syncthreads();
        float e = __expf(blog - gm);
        float z = e;                 // xor over bits 2-4 sums the 8 distinct o exactly once
        z += __shfl_xor(z, 4, 32);
        z += __shfl_xor(z, 8, 32);
        z += __shfl_xor(z, 16, 32);
        if ((t & 31) == 0) red[8 + wv] = z;
        __syncthreads();
        float Z = red[8];
        #pragma unroll
        for (int w = 1; w < 8; ++w) Z += red[8 + w];
        const float c = e / Z;
        #pragma unroll
        for (int j = 0; j < 8; ++j) acc[j] += c * u[j];
    }
    #pragma unroll
    for (int j = 0; j < 8; ++j)
        atomicAdd(&sacc[b * ODsz + od0 + j], acc[j]);
}

// ------------------------------------------------------------------ launch
extern "C" void kernel_launch(void* const* d_in, const int* in_sizes, int n_in,
                              void* d_out, int out_size, void* d_ws, size_t ws_size,
                              hipStream_t stream) {
    (void)in_sizes; (void)n_in; (void)out_size; (void)ws_size;
    const float* x    = (const float*)d_in[0];   // (16,2048,16)
    const float* W    = (const float*)d_in[1];   // (1,2048,64,32,16)
    const float* bias = (const float*)d_in[2];   // (1,1,64,32,1) -> 2048
    float* out = (float*)d_out;                  // (16,64,32)

    char*  ws   = (char*)d_ws;
    float* uhat = (float*)ws;                                   // 16*2048*2048 f32 = 256 MB
    float* sacc = (float*)(ws + (size_t)Bsz * Isz * ODsz * 4);  // 32768 f32
    float* vbuf = sacc + Bsz * ODsz;                            // 32768 f32
    float* b1   = vbuf + Bsz * ODsz;                            // 16*2048*64 f32 = 8 MB

    const int nS = Bsz * ODsz;                  // 32768
    zero_kernel<<<nS / 256, 256, 0, stream>>>(sacc, nS);

    // u_hat GEMM + fused sum_i (iter-0 uniform routing)
    uhat_gemm_kernel<<<dim3(ODsz / 512, Isz / IBLK), 256, 0, stream>>>(x, W, uhat, sacc);
    squash_kernel<<<nS / 256, 256, 0, stream>>>(sacc, bias, vbuf, 1.0f / (float)Osz);

    // iteration 1: b1 = u.v0, c1 = softmax, s1 = sum c1*u
    routing_pass_kernel<<<dim3(Isz / ICHUNK, Bsz), 256, 0, stream>>>(
        uhat, vbuf, nullptr, b1, sacc, 1);
    squash_kernel<<<nS / 256, 256, 0, stream>>>(sacc, bias, vbuf, 1.0f);

    // iteration 2: b2 = b1 + u.v1, c2 = softmax, s2 = sum c2*u
    routing_pass_kernel<<<dim3(Isz / ICHUNK, Bsz), 256, 0, stream>>>(
        uhat, vbuf, b1, nullptr, sacc, 2);
    squash_kernel<<<nS / 256, 256, 0, stream>>>(sacc, bias, out, 1.0f);
}